// QuantumNeuralNetwork_3710851743873
// MI455X (gfx1250) — compile-verified
//
#include <hip/hip_runtime.h>

#define NQ      14
#define DIM     16384
#define NLAYER  16
#define BATCH   512
#define TPB     512
#define EPT     (DIM / TPB)      // 32 elements per thread

typedef __attribute__((ext_vector_type(2))) float v2f;
typedef __attribute__((ext_vector_type(8))) float v8f;
typedef __attribute__((ext_vector_type(4))) int   v4i;

// ---- LDS layout (bytes) ----
#define SMEM_S      (DIM * 8)          // state: 16384 x float2 (interleaved re,im)
#define SMEM_CH     (128 * 8)          // RZ phase table, high 7 bits
#define SMEM_CL     (128 * 8)          // RZ phase table, low 7 bits
#define SMEM_G      (3 * 256 * 4)      // three 16x16 RY group matrices
#define SMEM_G4     (16 * 4)           // 4x4 matrix for qubits 12,13
#define SMEM_RED    (TPB * 4)          // reduction scratch
#define SMEM_TOTAL  (SMEM_S + SMEM_CH + SMEM_CL + SMEM_G + SMEM_G4 + SMEM_RED)

#if __has_builtin(__builtin_amdgcn_global_load_async_to_lds_b128)
#define HAVE_ASYNC_LDS 1
typedef __attribute__((address_space(1))) v4i* gptr_v4i;   // global-AS int4*
typedef __attribute__((address_space(3))) v4i* lptr_v4i;   // LDS-AS int4*
#endif

__device__ __forceinline__ float2 cmul(float2 a, float2 b) {
  return make_float2(a.x * b.x - a.y * b.y, a.x * b.y + a.y * b.x);
}

// Composed permutation of the ring of CNOTs (0,1),(1,2),...,(12,13),(13,0).
// dest bit (13-w) = XOR of source bits B0..Bw (w=1..13); dest bit 13 = parity ^ B0.
__device__ __forceinline__ int perm_cnot_ring(int k) {
  int y = k;
  y ^= y >> 1; y ^= y >> 2; y ^= y >> 4; y ^= y >> 8;
  int f0 = (y ^ (k >> 13)) & 1;
  return (y & 0x1FFF) | (f0 << 13);
}

__device__ __forceinline__ void block_reduce_store(float* red, int tid, float val, float* dst) {
  red[tid] = val;
  __syncthreads();
  #pragma unroll
  for (int s = TPB / 2; s > 0; s >>= 1) {
    if (tid < s) red[tid] += red[tid + s];
    __syncthreads();
  }
  if (tid == 0) *dst = red[0];
  __syncthreads();
}

// Apply one 16x16 RY group matrix (qubits at bit positions [LO, LO+3]) to the
// whole state via V_WMMA_F32_16X16X4_F32.  If FUSE_RZ, the RZ1 diagonal phase
// is applied to each element as it is loaded (only valid for LO==10, where the
// group covers the index MSBs and every element is touched exactly once).
template<int LO, bool FUSE_RZ>
__device__ __forceinline__ void ry_group_pass(float2* S, const float* Gm,
                                              const float2* CH, const float2* CL,
                                              int tid) {
  static_assert(!FUSE_RZ || LO == 10, "RZ fusion only valid for the MSB group");
  const int lane = tid & 31;
  const int wid  = tid >> 5;        // 16 waves
  const int hi   = lane >> 4;       // half-wave select
  const int n    = lane & 15;       // = A row M and B/D column N
  const int lomask = (1 << LO) - 1;

  // A operand: 16x4 slices of G.  VGPR0=K(0|2), VGPR1=K(1|3) per half-wave.
  v2f a[4];
  #pragma unroll
  for (int j = 0; j < 4; ++j) {
    a[j][0] = Gm[n * 16 + 4 * j + 2 * hi];
    a[j][1] = Gm[n * 16 + 4 * j + 2 * hi + 1];
  }
  #pragma unroll
  for (int it = 0; it < 4; ++it) {
    const int col  = (wid * 4 + it) * 16 + n;      // 64 column tiles, 4 per wave
    const int base = ((col >> LO) << (LO + 4)) | (col & lomask);
    float2 phL = make_float2(1.f, 0.f);
    if constexpr (FUSE_RZ) phL = CL[col & 127];
    // B operand: rows K of the state tile (VGPR v: K = 4j+v+2*hi)
    v2f bre[4], bim[4];
    #pragma unroll
    for (int j = 0; j < 4; ++j) {
      const int g0 = 4 * j + 2 * hi;
      float2 e0 = S[base | (g0 << LO)];
      float2 e1 = S[base | ((g0 + 1) << LO)];
      if constexpr (FUSE_RZ) {
        e0 = cmul(e0, cmul(CH[(g0 << 3) | (col >> 7)], phL));
        e1 = cmul(e1, cmul(CH[((g0 + 1) << 3) | (col >> 7)], phL));
      }
      bre[j][0] = e0.x; bre[j][1] = e1.x;
      bim[j][0] = e0.y; bim[j][1] = e1.y;
    }
    v8f cre = {0.f, 0.f, 0.f, 0.f, 0.f, 0.f, 0.f, 0.f};
    v8f cim = cre;
    #pragma unroll
    for (int j = 0; j < 4; ++j)
      cre = __builtin_amdgcn_wmma_f32_16x16x4_f32(false, a[j], false, bre[j],
                                                  (short)0, cre, false, false);
    #pragma unroll
    for (int j = 0; j < 4; ++j)
      cim = __builtin_amdgcn_wmma_f32_16x16x4_f32(false, a[j], false, bim[j],
                                                  (short)0, cim, false, false);
    // D: VGPR v -> row M = v + 8*hi, same column
    #pragma unroll
    for (int v = 0; v < 8; ++v)
      S[base | ((v + 8 * hi) << LO)] = make_float2(cre[v], cim[v]);
  }
}

__global__ __launch_bounds__(TPB) void qnn_kernel(const float2* __restrict__ z,
                                                  const float* __restrict__ theta,
                                                  float* __restrict__ out) {
  extern __shared__ char smem[];
  float2* S   = (float2*)smem;
  float2* CH  = (float2*)(smem + SMEM_S);
  float2* CL  = (float2*)(smem + SMEM_S + SMEM_CH);
  float*  G   = (float*)(smem + SMEM_S + SMEM_CH + SMEM_CL);
  float*  G4  = G + 3 * 256;
  float*  red = G4 + 16;

  const int tid = threadIdx.x;
  const int b   = blockIdx.x;

  __builtin_prefetch((const void*)theta, 0, 1);   // global_prefetch_b8

  // ---- load state (one pass over HBM), async direct-to-LDS if available ----
  const float2* zb = z + (size_t)b * DIM;
#if defined(HAVE_ASYNC_LDS)
  {
    const char* gsrc = (const char*)zb;
    char*       ldst = (char*)S;
    #pragma unroll
    for (int i = 0; i < EPT / 2; ++i) {            // 16 x B128 per thread
      size_t off = (size_t)(tid + i * TPB) * 16;
      gptr_v4i g = (gptr_v4i)(uintptr_t)(gsrc + off);
      lptr_v4i l = (lptr_v4i)(uint32_t)(uintptr_t)(ldst + off);
      __builtin_amdgcn_global_load_async_to_lds_b128(g, l, 0, 0);
    }
#if __has_builtin(__builtin_amdgcn_s_wait_asynccnt)
    __builtin_amdgcn_s_wait_asynccnt(0);
#else
    asm volatile("s_wait_asynccnt 0" ::: "memory");
#endif
  }
#else
  #pragma unroll
  for (int i = 0; i < EPT; ++i) S[tid + i * TPB] = zb[tid + i * TPB];
#endif
  __syncthreads();

  for (int l = 0; l < NLAYER; ++l) {
    const float* th  = theta + l * 3 * NQ;
    const float* rz1 = th;
    const float* ry  = th + NQ;
    const float* rz2 = th + 2 * NQ;

    // ---------- per-layer prep: RY group matrices + RZ1 phase tables ----------
    if (tid < 256) {
      int m = tid >> 4, n = tid & 15;
      for (int grp = 0; grp < 3; ++grp) {
        float prod = 1.f;
        #pragma unroll
        for (int j = 0; j < 4; ++j) {
          float s, c; __sincosf(ry[4 * grp + j] * 0.5f, &s, &c);
          int ma = (m >> (3 - j)) & 1, nb = (n >> (3 - j)) & 1;
          prod *= ma ? (nb ? c : s) : (nb ? -s : c);   // R = [[c,-s],[s,c]]
        }
        G[grp * 256 + m * 16 + n] = prod;
      }
      if (tid < 16) {
        int mm = tid >> 2, nn = tid & 3;
        float prod = 1.f;
        #pragma unroll
        for (int j = 0; j < 2; ++j) {
          float s, c; __sincosf(ry[12 + j] * 0.5f, &s, &c);
          int ma = (mm >> (1 - j)) & 1, nb = (nn >> (1 - j)) & 1;
          prod *= ma ? (nb ? c : s) : (nb ? -s : c);
        }
        G4[tid] = prod;
      }
    } else if (tid < 384) {          // CH table (qubits 0..6 -> k bits 13..7)
      int i = tid - 256;
      float ang = 0.f;
      for (int w = 0; w < NQ; ++w) ang -= 0.5f * rz1[w];   // fold base into CH
      for (int j = 0; j < 7; ++j) if ((i >> j) & 1) ang += rz1[6 - j];
      float s, c; __sincosf(ang, &s, &c);
      CH[i] = make_float2(c, s);
    } else {                          // CL table (qubits 7..13 -> k bits 6..0)
      int i = tid - 384;
      float ang = 0.f;
      for (int j = 0; j < 7; ++j) if ((i >> j) & 1) ang += rz1[13 - j];
      float s, c; __sincosf(ang, &s, &c);
      CL[i] = make_float2(c, s);
    }
    __syncthreads();

    // ---------- RY WMMA passes; RZ1 fused into the first (MSB) group ----------
    ry_group_pass<10, true >(S, G,       CH, CL, tid);
    __syncthreads();
    ry_group_pass<6,  false>(S, G + 256, CH, CL, tid);
    __syncthreads();
    ry_group_pass<2,  false>(S, G + 512, CH, CL, tid);
    __syncthreads();

    // ---------- rebuild phase tables for RZ2 ----------
    if (tid < 128) {
      float ang = 0.f;
      for (int w = 0; w < NQ; ++w) ang -= 0.5f * rz2[w];
      for (int j = 0; j < 7; ++j) if ((tid >> j) & 1) ang += rz2[6 - j];
      float s, c; __sincosf(ang, &s, &c);
      CH[tid] = make_float2(c, s);
    } else if (tid < 256) {
      int i = tid - 128;
      float ang = 0.f;
      for (int j = 0; j < 7; ++j) if ((i >> j) & 1) ang += rz2[13 - j];
      float s, c; __sincosf(ang, &s, &c);
      CL[i] = make_float2(c, s);
    }
    __syncthreads();

    // ---- fused: RY(qubits 12,13) 4x4 + RZ2 phase + CNOT-ring permutation ----
    {
      float2 stage[EPT];
      #pragma unroll
      for (int i = 0; i < 8; ++i) {
        int c = tid + i * TPB;          // 4096 quads of 4 consecutive amplitudes
        int bi = c << 2;
        float2 x0 = S[bi], x1 = S[bi + 1], x2 = S[bi + 2], x3 = S[bi + 3];
        float2 y[4];
        #pragma unroll
        for (int m = 0; m < 4; ++m) {
          y[m].x = G4[m*4+0]*x0.x + G4[m*4+1]*x1.x + G4[m*4+2]*x2.x + G4[m*4+3]*x3.x;
          y[m].y = G4[m*4+0]*x0.y + G4[m*4+1]*x1.y + G4[m*4+2]*x2.y + G4[m*4+3]*x3.y;
        }
        float2 chv = CH[c >> 5];        // (c*4+g) >> 7 == c >> 5 for g in 0..3
        #pragma unroll
        for (int g = 0; g < 4; ++g) {
          float2 ph = cmul(chv, CL[((c & 31) << 2) | g]);
          stage[i * 4 + g] = cmul(y[g], ph);
        }
      }
      __syncthreads();
      #pragma unroll
      for (int i = 0; i < 8; ++i) {
        int c = tid + i * TPB;
        #pragma unroll
        for (int g = 0; g < 4; ++g) {
          int k = (c << 2) | g;
          S[perm_cnot_ring(k)] = stage[i * 4 + g];
        }
      }
      __syncthreads();
    }
  }

  // ---------- expectations: <Z0>, <Z1>, 2Re<X0-cross>, 2Re<X1-cross>, 2Im<X0-cross> ----------
  float ez0 = 0.f, ez1 = 0.f, ex0 = 0.f, ex1 = 0.f, ey0 = 0.f;
  #pragma unroll
  for (int i = 0; i < EPT; ++i) {
    int k = tid + i * TPB;
    float2 s = S[k];
    float p = s.x * s.x + s.y * s.y;
    ez0 += ((k >> 13) & 1) ? -p : p;
    ez1 += ((k >> 12) & 1) ? -p : p;
  }
  #pragma unroll
  for (int i = 0; i < 16; ++i) {
    int j = tid + i * TPB;                 // 8192 pairs
    float2 a0 = S[j], a1 = S[j + 8192];    // qubit 0 pair (bit 13)
    ex0 += a0.x * a1.x + a0.y * a1.y;      // Re(conj(p0)*p1)
    ey0 += a0.x * a1.y - a0.y * a1.x;      // Im(conj(p0)*p1)
    int c0i = ((j >> 12) << 13) | (j & 0xFFF);   // qubit 1 pair (bit 12)
    float2 c0 = S[c0i], c1 = S[c0i | (1 << 12)];
    ex1 += c0.x * c1.x + c0.y * c1.y;
  }
  ex0 *= 2.f; ex1 *= 2.f; ey0 *= 2.f;

  float* row = out + 1 + 5 * b;
  __syncthreads();
  block_reduce_store(red, tid, ez0, row + 0);
  block_reduce_store(red, tid, ez1, row + 1);
  block_reduce_store(red, tid, ex0, row + 2);
  block_reduce_store(red, tid, ex1, row + 3);
  block_reduce_store(red, tid, ey0, row + 4);
}

// ---- log-softmax NLL over the 512x5 output rows ----
__global__ __launch_bounds__(TPB) void loss_kernel(const int* __restrict__ y,
                                                   float* __restrict__ out) {
  __shared__ float red[TPB];
  int t = threadIdx.x;
  const float* row = out + 1 + 5 * t;
  float m = row[0];
  #pragma unroll
  for (int c = 1; c < 5; ++c) m = fmaxf(m, row[c]);
  float sum = 0.f;
  #pragma unroll
  for (int c = 0; c < 5; ++c) sum += __expf(row[c] - m);
  float lse = m + __logf(sum);
  float val = lse - row[y[t]];             // -logp[y]
  red[t] = val;
  __syncthreads();
  #pragma unroll
  for (int s = TPB / 2; s > 0; s >>= 1) {
    if (t < s) red[t] += red[t + s];
    __syncthreads();
  }
  if (t == 0) out[0] = red[0] / (float)BATCH;
}

extern "C" void kernel_launch(void* const* d_in, const int* in_sizes, int n_in,
                              void* d_out, int out_size, void* d_ws, size_t ws_size,
                              hipStream_t stream) {
  (void)in_sizes; (void)n_in; (void)out_size; (void)d_ws; (void)ws_size;
  const float2* z     = (const float2*)d_in[0];
  const float*  theta = (const float*)d_in[1];
  const int*    y     = (const int*)d_in[2];
  float* out = (float*)d_out;

  (void)hipFuncSetAttribute(reinterpret_cast<const void*>(qnn_kernel),
                            hipFuncAttributeMaxDynamicSharedMemorySize, SMEM_TOTAL);
  qnn_kernel<<<BATCH, TPB, SMEM_TOTAL, stream>>>(z, theta, out);
  loss_kernel<<<1, TPB, 0, stream>>>(y, out);
}